// get_model_42219528519805
// MI455X (gfx1250) — compile-verified
//
#include <hip/hip_runtime.h>
#include <hip/hip_bf16.h>
#include <math.h>

typedef __attribute__((ext_vector_type(2))) float v2f;
typedef __attribute__((ext_vector_type(8))) float v8f;

#define Bc   2
#define Nn   10000
#define Hh   8
#define HCc  32
#define UVc  25
#define Dd   5
#define HWc  400
#define Lc   50000      // Dd * Nn
#define SEGc 125        // UVc * Dd

static __device__ __forceinline__ v8f wmma_f32(v2f a, v2f b, v8f c) {
  // D = A(16x4, f32) * B(4x16, f32) + C(16x16, f32)
  return __builtin_amdgcn_wmma_f32_16x16x4_f32(
      /*neg_a=*/false, a, /*neg_b=*/false, b,
      /*c_mod=*/(short)0, c, /*reuse_a=*/false, /*reuse_b=*/false);
}

// ---------------------------------------------------------------------------
// Kernel 1: per (b,d,w,h) compute S = (Q · K^T) / sqrt(HC), 25x25 (padded to
// 32x32x32), via v_wmma_f32_16x16x4_f32. One wave per head, 4 waves/block.
// Writes AW1[((b*L + (d*25+u)*400+w)*H + h)*25 + v] = S[u][v]*scale.
// ---------------------------------------------------------------------------
__global__ __launch_bounds__(128) void qk_kernel(const float* __restrict__ qkv,
                                                 const int* __restrict__ attn_idx,
                                                 float* __restrict__ aw1) {
  __shared__ __align__(16) float sQ[4][32][34];
  __shared__ __align__(16) float sK[4][32][34];

  const int gid   = blockIdx.x;
  const int grp   = gid >> 1;
  const int hbase = (gid & 1) * 4;
  const int b     = grp / (Dd * HWc);
  const int rem   = grp % (Dd * HWc);
  const int d     = rem / HWc;
  const int w     = rem % HWc;
  const int wave  = threadIdx.x >> 5;
  const int lane  = threadIdx.x & 31;
  const int h     = hbase + wave;

  // Stage Q and K tiles (coalesced: lane = channel). Rows >= 25 zero-padded.
  for (int u = 0; u < 32; ++u) {
    float qv = 0.f, kv = 0.f;
    if (u < UVc) {
      const int l = (d * UVc + u) * HWc + w;
      const int r = attn_idx[l];
      const size_t qoff = ((size_t)((0 * Bc + b) * Nn + r) * Hh + h) * HCc + lane;
      const size_t koff = ((size_t)((1 * Bc + b) * Nn + r) * Hh + h) * HCc + lane;
      qv = qkv[qoff];
      kv = qkv[koff];
    }
    sQ[wave][u][lane] = qv;
    sK[wave][u][lane] = kv;
  }
  __syncthreads();

  const int mn = lane & 15;   // M (A) / N (B) position within tile
  const int kh = lane >> 4;   // K-half select
  v8f c00 = {}; v8f c01 = {}; v8f c10 = {}; v8f c11 = {};
#pragma unroll
  for (int s = 0; s < 8; ++s) {
    const int c0 = kh * 2 + s * 4;   // even -> 8B-aligned LDS reads
    v2f a0 = *(const v2f*)&sQ[wave][mn][c0];
    v2f a1 = *(const v2f*)&sQ[wave][mn + 16][c0];
    v2f b0 = *(const v2f*)&sK[wave][mn][c0];        // B[k][n] = K[n][k]
    v2f b1 = *(const v2f*)&sK[wave][mn + 16][c0];
    c00 = wmma_f32(a0, b0, c00);
    c01 = wmma_f32(a0, b1, c01);
    c10 = wmma_f32(a1, b0, c10);
    c11 = wmma_f32(a1, b1, c11);
  }

  const float scale = 0.17677669529663687f;  // 1/sqrt(32)
#pragma unroll
  for (int j = 0; j < 8; ++j) {
    const int mr = j + 8 * kh;
    {
      const int row = mr;           // tile M=0..15
      if (row < UVc) {
        const size_t base =
            ((size_t)(b * Lc + (d * UVc + row) * HWc + w) * Hh + h) * UVc;
        aw1[base + mn] = c00[j] * scale;                       // cols 0..15 (<25)
        if (mn + 16 < UVc) aw1[base + mn + 16] = c01[j] * scale;
      }
    }
    {
      const int row = mr + 16;      // tile M=16..31 (rows 16..24 valid)
      if (row < UVc) {
        const size_t base =
            ((size_t)(b * Lc + (d * UVc + row) * HWc + w) * Hh + h) * UVc;
        aw1[base + mn] = c10[j] * scale;
        if (mn + 16 < UVc) aw1[base + mn + 16] = c11[j] * scale;
      }
    }
  }
}

// ---------------------------------------------------------------------------
// Kernel 2: per (b,p,h): gather 125 scores via attn_inv (slot = v*5 + d),
// masked softmax, disp = sum(prob * disp_map), and scatter probs into
// AWR[((b*L + d*N + p)*H + h)*25 + v] so kernel 3 can gather via attn_idx2.
// ---------------------------------------------------------------------------
__global__ __launch_bounds__(256) void softmax_kernel(
    const float* __restrict__ aw1, const float* __restrict__ disp_map,
    const int* __restrict__ mask, const int* __restrict__ attn_inv,
    float* __restrict__ awr, float* __restrict__ dout) {
  const int tid = blockIdx.x * blockDim.x + threadIdx.x;
  if (tid >= Bc * Nn * Hh) return;
  const int h = tid % Hh;
  const int p = (tid / Hh) % Nn;
  const int b = tid / (Hh * Nn);

  float vals[SEGc];
#pragma unroll
  for (int d = 0; d < Dd; ++d) {
    const int l = attn_inv[d * Nn + p];
    const size_t base = ((size_t)(b * Lc + l) * Hh + h) * UVc;
#pragma unroll
    for (int v = 0; v < UVc; ++v) vals[v * Dd + d] = aw1[base + v];
  }

  float mx = -INFINITY;
#pragma unroll
  for (int s = 0; s < SEGc; ++s)
    if (mask[p * SEGc + s]) mx = fmaxf(mx, vals[s]);

  float sum = 0.f, dn = 0.f;
#pragma unroll
  for (int s = 0; s < SEGc; ++s) {
    const float e = mask[p * SEGc + s] ? __expf(vals[s] - mx) : 0.f;
    vals[s] = e;
    sum += e;
    dn = fmaf(e, disp_map[s], dn);
  }
  const float rinv = 1.f / sum;

  // disp output lives after attn_out (B*N*H*HC floats) in d_out
  dout[(size_t)Bc * Nn * (Hh * HCc) + ((size_t)(b * Nn + p)) * Hh + h] = dn * rinv;

#pragma unroll
  for (int d = 0; d < Dd; ++d) {
    const size_t base = ((size_t)(b * Lc + d * Nn + p) * Hh + h) * UVc;
#pragma unroll
    for (int v = 0; v < UVc; ++v) awr[base + v] = vals[v * Dd + d] * rinv;
  }
}

// ---------------------------------------------------------------------------
// Kernel 3: per (b,d,w,h): O = A(25x25, gathered via attn_idx2) · V(25x32,
// gathered via attn_idx), padded to 32x32x32 WMMA; atomicAdd rows into
// attn_out[b, attn_idx[(d*25+u)*400+w], h*32 + c].
// ---------------------------------------------------------------------------
__global__ __launch_bounds__(128) void av_kernel(const float* __restrict__ qkv,
                                                 const int* __restrict__ attn_idx,
                                                 const int* __restrict__ attn_idx2,
                                                 const float* __restrict__ awr,
                                                 float* __restrict__ attn_out) {
  __shared__ __align__(16) float sA[4][32][34];
  __shared__ __align__(16) float sV[4][32][34];
  __shared__ int sR[4][UVc];

  const int gid   = blockIdx.x;
  const int grp   = gid >> 1;
  const int hbase = (gid & 1) * 4;
  const int b     = grp / (Dd * HWc);
  const int rem   = grp % (Dd * HWc);
  const int d     = rem / HWc;
  const int w     = rem % HWc;
  const int wave  = threadIdx.x >> 5;
  const int lane  = threadIdx.x & 31;
  const int h     = hbase + wave;

  for (int u = 0; u < 32; ++u) {
    float av = 0.f, vv = 0.f;
    if (u < UVc) {
      const int j    = (d * UVc + u) * HWc + w;
      const int rsrc = attn_idx[j];
      if (lane == 0) sR[wave][u] = rsrc;  // destination row for output scatter
      const int l2 = attn_idx2[j];
      if (lane < UVc)
        av = awr[((size_t)(b * Lc + l2) * Hh + h) * UVc + lane];
      vv = qkv[((size_t)((2 * Bc + b) * Nn + rsrc) * Hh + h) * HCc + lane];
    }
    sA[wave][u][lane] = av;   // zero-padded rows & cols (K>=25 is zero)
    sV[wave][u][lane] = vv;   // zero-padded rows avoid 0*garbage NaN
  }
  __syncthreads();

  const int mn = lane & 15;
  const int kh = lane >> 4;
  v8f c00 = {}; v8f c01 = {}; v8f c10 = {}; v8f c11 = {};
#pragma unroll
  for (int s = 0; s < 8; ++s) {
    const int c0 = kh * 2 + s * 4;
    v2f a0 = *(const v2f*)&sA[wave][mn][c0];
    v2f a1 = *(const v2f*)&sA[wave][mn + 16][c0];
    v2f b0, b1;                         // B[k][n] = V[k][n]
    b0.x = sV[wave][c0][mn];     b0.y = sV[wave][c0 + 1][mn];
    b1.x = sV[wave][c0][mn + 16]; b1.y = sV[wave][c0 + 1][mn + 16];
    c00 = wmma_f32(a0, b0, c00);
    c01 = wmma_f32(a0, b1, c01);
    c10 = wmma_f32(a1, b0, c10);
    c11 = wmma_f32(a1, b1, c11);
  }

#pragma unroll
  for (int j = 0; j < 8; ++j) {
    const int mr = j + 8 * kh;
    {
      const int row = mr;
      if (row < UVc) {
        const size_t base =
            ((size_t)(b * Nn + sR[wave][row])) * (Hh * HCc) + (size_t)h * HCc;
        atomicAdd(&attn_out[base + mn], c00[j]);
        atomicAdd(&attn_out[base + mn + 16], c01[j]);   // all 32 cols valid
      }
    }
    {
      const int row = mr + 16;
      if (row < UVc) {
        const size_t base =
            ((size_t)(b * Nn + sR[wave][row])) * (Hh * HCc) + (size_t)h * HCc;
        atomicAdd(&attn_out[base + mn], c10[j]);
        atomicAdd(&attn_out[base + mn + 16], c11[j]);
      }
    }
  }
}

// ---------------------------------------------------------------------------
extern "C" void kernel_launch(void* const* d_in, const int* in_sizes, int n_in,
                              void* d_out, int out_size, void* d_ws, size_t ws_size,
                              hipStream_t stream) {
  (void)in_sizes; (void)n_in; (void)out_size; (void)ws_size;
  const float* qkv       = (const float*)d_in[0];
  const float* disp_map  = (const float*)d_in[1];
  const int*   mask      = (const int*)d_in[2];
  const int*   attn_idx  = (const int*)d_in[3];
  const int*   attn_idx2 = (const int*)d_in[4];
  const int*   attn_inv  = (const int*)d_in[5];

  float* aw1 = (float*)d_ws;                          // B*L*H*25 f32 = 80 MB
  float* awr = aw1 + (size_t)Bc * Lc * Hh * UVc;      // another 80 MB
  float* out = (float*)d_out;

  // Zero attn_out region (scatter-add target); disp region fully overwritten.
  hipMemsetAsync(d_out, 0, (size_t)Bc * Nn * Hh * HCc * sizeof(float), stream);

  qk_kernel<<<dim3(Bc * Dd * HWc * 2), dim3(128), 0, stream>>>(qkv, attn_idx, aw1);
  softmax_kernel<<<dim3((Bc * Nn * Hh + 255) / 256), dim3(256), 0, stream>>>(
      aw1, disp_map, mask, attn_inv, awr, out);
  av_kernel<<<dim3(Bc * Dd * HWc * 2), dim3(128), 0, stream>>>(
      qkv, attn_idx, attn_idx2, awr, out);
}